// CascadeFasterRCNN_64493228917074
// MI455X (gfx1250) — compile-verified
//
#include <hip/hip_runtime.h>
#include <stdint.h>

#define NLVL   4
#define NPRE   2000
#define NPOST  1000
#define KCAND  (NLVL * NPRE)   // 8000
#define KPAD   8192
#define MWORDS (KCAND / 32)    // 250
#define NMS_THR 0.7f
#define MIN_WH  1.0f

// ---- order-preserving float <-> uint32 key (descending score == descending key)
__device__ __forceinline__ unsigned fkey(float f) {
  unsigned u = __float_as_uint(f);
  return (u & 0x80000000u) ? ~u : (u | 0x80000000u);
}
__device__ __forceinline__ float fdec(unsigned k) {
  unsigned u = (k & 0x80000000u) ? (k & 0x7FFFFFFFu) : ~k;
  return __uint_as_float(u);
}

// ---- 1. zero the meta + candidate region of the workspace
__global__ void k_init(unsigned* p, int n) {
  for (int i = blockIdx.x * blockDim.x + threadIdx.x; i < n; i += gridDim.x * blockDim.x)
    p[i] = 0u;
}

// ---- 2. score keys: softmax(logits)[:,1] is monotone in (l1 - l0)
__global__ void k_keys(const float* __restrict__ logits, unsigned* __restrict__ keys, int N) {
  int i = blockIdx.x * blockDim.x + threadIdx.x;
  if (i < N) keys[i] = fkey(logits[2 * i + 1] - logits[2 * i + 0]);
}

// ---- 3. exact per-level radix select of the NPRE-th largest key (3 rounds: 11/11/10 bits)
__global__ __launch_bounds__(1024) void k_select(const unsigned* __restrict__ keys,
                                                 const int* __restrict__ levels, int N,
                                                 unsigned* T, unsigned* cntG, unsigned* needE) {
  int lvl = blockIdx.x;
  __shared__ unsigned hist[2048];
  __shared__ unsigned s_prefix, s_rank, s_above;
  if (threadIdx.x == 0) { s_prefix = 0u; s_rank = NPRE; s_above = 0u; }
  const int shifts[3] = {21, 10, 0};
  const int bins[3]   = {2048, 2048, 1024};
  for (int r = 0; r < 3; ++r) {
    for (int b = threadIdx.x; b < bins[r]; b += blockDim.x) hist[b] = 0u;
    __syncthreads();
    unsigned prefix = s_prefix;
    unsigned pmask  = (r == 0) ? 0u : (0xFFFFFFFFu << shifts[r - 1]);
    for (int i = threadIdx.x; i < N; i += blockDim.x) {
      if (levels[i] == lvl) {
        unsigned k = keys[i];
        if ((k & pmask) == prefix) {
          unsigned d = (k >> shifts[r]) & (unsigned)(bins[r] - 1);
          atomicAdd(&hist[d], 1u);
        }
      }
    }
    __syncthreads();
    if (threadIdx.x == 0) {
      unsigned rank = s_rank, cum = 0u; int chosen = 0;
      for (int d = bins[r] - 1; d >= 0; --d) {
        unsigned c = hist[d];
        if (cum + c >= rank) { chosen = d; s_above += cum; s_rank = rank - cum; break; }
        cum += c;
      }
      s_prefix |= ((unsigned)chosen << shifts[r]);
    }
    __syncthreads();
  }
  if (threadIdx.x == 0) {
    T[lvl] = s_prefix; cntG[lvl] = s_above; needE[lvl] = NPRE - s_above;
  }
}

// ---- 4. compact per-level candidates into 4*2000 slots
__global__ void k_compact(const unsigned* __restrict__ keys, const int* __restrict__ levels, int N,
                          const unsigned* T, const unsigned* cntG, const unsigned* needE,
                          unsigned* ctrG, unsigned* ctrE,
                          unsigned* candKey, unsigned* candIdx) {
  int i = blockIdx.x * blockDim.x + threadIdx.x;
  if (i >= N) return;
  int lvl = levels[i];
  unsigned k = keys[i], t = T[lvl];
  if (k > t) {
    unsigned p = atomicAdd(&ctrG[lvl], 1u);
    if (p < (unsigned)NPRE) { unsigned s = lvl * NPRE + p; candKey[s] = k; candIdx[s] = (unsigned)i; }
  } else if (k == t) {
    unsigned p = atomicAdd(&ctrE[lvl], 1u);
    if (p < needE[lvl]) {
      unsigned s = lvl * NPRE + cntG[lvl] + p;
      if (s < (unsigned)((lvl + 1) * NPRE)) { candKey[s] = k; candIdx[s] = (unsigned)i; }
    }
  }
}

// ---- 5. single-block bitonic sort of 8192 (key,slot) pairs, descending by key
__global__ __launch_bounds__(1024) void k_sort(const unsigned* __restrict__ candKey,
                                               unsigned* __restrict__ sortedSlot) {
  __shared__ unsigned sk[KPAD];
  __shared__ unsigned sv[KPAD];
  for (int t = threadIdx.x; t < KPAD; t += blockDim.x) {
    if (t < KCAND) { sk[t] = candKey[t]; sv[t] = (unsigned)t; }
    else           { sk[t] = 0u;         sv[t] = 0xFFFFFFFFu; }
  }
  __syncthreads();
  for (int k = 2; k <= KPAD; k <<= 1) {
    for (int j = k >> 1; j > 0; j >>= 1) {
      for (int i = threadIdx.x; i < KPAD; i += blockDim.x) {
        int ij = i ^ j;
        if (ij > i) {
          unsigned ka = sk[i], kb = sk[ij], va = sv[i], vb = sv[ij];
          bool aFirst = (ka > kb) || (ka == kb && va < vb); // descending, tie: lower slot first
          bool descBlock = ((i & k) == 0);
          if (descBlock ? !aFirst : aFirst) {
            sk[i] = kb; sk[ij] = ka; sv[i] = vb; sv[ij] = va;
          }
        }
      }
      __syncthreads();
    }
  }
  for (int t = threadIdx.x; t < KPAD; t += blockDim.x) sortedSlot[t] = sv[t];
}

// ---- 6. sep = max over all candidate roi coords (orderable-uint atomicMax)
__global__ void k_sepmax(const float* __restrict__ rois, const unsigned* __restrict__ candIdx,
                         unsigned* sepmax) {
  int t = blockIdx.x * blockDim.x + threadIdx.x;
  if (t >= KCAND) return;
  unsigned idx = candIdx[t];
  unsigned m = 0u;
  for (int c = 0; c < 4; ++c) m = max(m, fkey(rois[idx * 4 + c]));
  atomicMax(sepmax, m);
}

// ---- 7. build level-shifted boxes in sorted order + sorted anchor idx / level
__global__ void k_shift(const float* __restrict__ rois, const unsigned* __restrict__ candIdx,
                        const unsigned* __restrict__ sortedSlot, const unsigned* sepmax,
                        float* __restrict__ boxes, unsigned* __restrict__ sortIdx,
                        int* __restrict__ sortLvl) {
  int s = blockIdx.x * blockDim.x + threadIdx.x;
  if (s >= KCAND) return;
  float sep = fdec(*sepmax) + 1.0f;
  unsigned slot = sortedSlot[s];
  if (slot < (unsigned)KCAND) {
    unsigned idx = candIdx[slot];
    int lvl = (int)(slot / NPRE);
    float off = (float)lvl * sep;
    for (int c = 0; c < 4; ++c) boxes[s * 4 + c] = rois[idx * 4 + c] + off;
    sortIdx[s] = idx; sortLvl[s] = lvl;
  } else {
    float base = -1.0e8f - (float)s * 1000.0f; // isolated degenerate box
    boxes[s * 4 + 0] = base; boxes[s * 4 + 1] = base;
    boxes[s * 4 + 2] = base + 0.5f; boxes[s * 4 + 3] = base + 0.5f;
    sortIdx[s] = 0u; sortLvl[s] = -1;
  }
}

// ---- 8. IoU suppression bitmask: wave32 blocks, column boxes staged into LDS
//          via the CDNA5 async global->LDS engine (ASYNCcnt path)
__global__ __launch_bounds__(32) void k_mask(const float* __restrict__ boxes,
                                             unsigned* __restrict__ mask) {
  int bj = blockIdx.x, bi = blockIdx.y;
  int t = threadIdx.x;
  int i = bi * 32 + t;
  if (bj < bi) { mask[(size_t)i * MWORDS + bj] = 0u; return; }

  __shared__ float cb[32 * 4];
  {
    const float* src = boxes + (size_t)(bj * 32 + t) * 4;
    unsigned ldsAddr = (unsigned)(unsigned long long)(uintptr_t)(&cb[t * 4]);
    unsigned long long ga = (unsigned long long)(uintptr_t)src;
    // CDNA5: GLOBAL_LOAD_ASYNC_TO_LDS_B128  (16B per lane, tracked by ASYNCcnt)
    asm volatile("global_load_async_to_lds_b128 %0, %1, off"
                 :: "v"(ldsAddr), "v"(ga) : "memory");
    asm volatile("s_wait_asynccnt 0" ::: "memory");
  }
  __syncthreads(); // single wave: barrier is a NOP, kept for memory-ordering clarity

  float y1 = boxes[i * 4 + 0], x1 = boxes[i * 4 + 1];
  float y2 = boxes[i * 4 + 2], x2 = boxes[i * 4 + 3];
  float areaI = (y2 - y1) * (x2 - x1);

  unsigned bits = 0u;
  #pragma unroll 4
  for (int jj = 0; jj < 32; ++jj) {
    int j = bj * 32 + jj;
    if (j <= i) continue;
    float by1 = cb[jj * 4 + 0], bx1 = cb[jj * 4 + 1];
    float by2 = cb[jj * 4 + 2], bx2 = cb[jj * 4 + 3];
    float iy1 = fmaxf(y1, by1), ix1 = fmaxf(x1, bx1);
    float iy2 = fminf(y2, by2), ix2 = fminf(x2, bx2);
    float ih = fmaxf(iy2 - iy1, 0.0f), iw = fmaxf(ix2 - ix1, 0.0f);
    float inter = ih * iw;
    float areaJ = (by2 - by1) * (bx2 - bx1);
    float uni = areaI + areaJ - inter;
    float iou = inter / fmaxf(uni, 1e-9f);
    if (iou > NMS_THR) bits |= (1u << jj);
  }
  mask[(size_t)i * MWORDS + bj] = bits;
}

// ---- 9. greedy sequential reduce over the bitmask (one wave32, lockstep)
__global__ __launch_bounds__(32) void k_reduce(const unsigned* __restrict__ mask,
                                               unsigned* __restrict__ selOut,
                                               unsigned* __restrict__ nkeepOut) {
  __shared__ unsigned remv[MWORDS];
  __shared__ unsigned nsel;
  int t = threadIdx.x;
  for (int w = t; w < MWORDS; w += 32) remv[w] = 0u;
  if (t == 0) nsel = 0u;
  __syncthreads();
  for (int i = 0; i < KCAND; ++i) {
    unsigned rw = remv[i >> 5]; // uniform LDS broadcast read
    bool keep = !((rw >> (i & 31)) & 1u);
    if (keep) {
      if (t == 0) { if (nsel < (unsigned)NPOST) selOut[nsel] = (unsigned)i; nsel++; }
      const unsigned* row = mask + (size_t)i * MWORDS;
      for (int w = t; w < MWORDS; w += 32) remv[w] |= row[w];
    }
    __syncthreads();
  }
  if (t == 0) *nkeepOut = nsel;
}

// ---- 10. gather outputs: [logits 2000][rois 4000][levels 1000][valid 1000] (all f32)
__global__ void k_out(const float* __restrict__ logits, const float* __restrict__ rois,
                      const unsigned* __restrict__ sortIdx, const int* __restrict__ sortLvl,
                      const unsigned* __restrict__ sel, const unsigned* __restrict__ nkeep,
                      float* __restrict__ out) {
  int o = blockIdx.x * blockDim.x + threadIdx.x;
  if (o >= NPOST) return;
  unsigned nk = *nkeep;
  float l0 = 0.f, l1 = 0.f, r0 = 0.f, r1 = 0.f, r2 = 0.f, r3 = 0.f;
  float lv = -1.0f, vd = 0.0f;
  if ((unsigned)o < nk) {
    unsigned s = sel[o];
    unsigned idx = sortIdx[s];
    int lvl = sortLvl[s];
    float y1 = rois[idx * 4 + 0], x1 = rois[idx * 4 + 1];
    float y2 = rois[idx * 4 + 2], x2 = rois[idx * 4 + 3];
    if ((y2 - y1) >= MIN_WH && (x2 - x1) >= MIN_WH) {
      l0 = logits[idx * 2 + 0]; l1 = logits[idx * 2 + 1];
      r0 = y1; r1 = x1; r2 = y2; r3 = x2;
      lv = (float)lvl; vd = 1.0f;
    }
  }
  out[o * 2 + 0] = l0; out[o * 2 + 1] = l1;
  float* outR = out + 2 * NPOST;
  outR[o * 4 + 0] = r0; outR[o * 4 + 1] = r1; outR[o * 4 + 2] = r2; outR[o * 4 + 3] = r3;
  float* outL = out + 6 * NPOST; outL[o] = lv;
  float* outV = out + 7 * NPOST; outV[o] = vd;
}

extern "C" void kernel_launch(void* const* d_in, const int* in_sizes, int n_in,
                              void* d_out, int out_size, void* d_ws, size_t ws_size,
                              hipStream_t stream) {
  (void)n_in; (void)out_size; (void)ws_size;
  const float* logits = (const float*)d_in[0];
  const float* rois   = (const float*)d_in[1];
  const int*   levels = (const int*)d_in[2];
  int N = in_sizes[0] / 2;

  // workspace layout (u32 units)
  unsigned* W        = (unsigned*)d_ws;
  unsigned* keys     = W;                       // N
  unsigned* meta     = W + N;                   // 32
  unsigned* Tarr     = meta + 0;                // 4
  unsigned* cntG     = meta + 4;                // 4
  unsigned* needE    = meta + 8;                // 4
  unsigned* ctrG     = meta + 12;               // 4
  unsigned* ctrE     = meta + 16;               // 4
  unsigned* sepmax   = meta + 20;               // 1
  unsigned* nkeep    = meta + 21;               // 1
  unsigned* candKey  = meta + 32;               // KCAND
  unsigned* candIdx  = candKey + KCAND;         // KCAND
  unsigned* sortedSl = candIdx + KCAND;         // KPAD
  unsigned* sortIdx  = sortedSl + KPAD;         // KCAND
  int*      sortLvl  = (int*)(sortIdx + KCAND); // KCAND
  unsigned* sel      = (unsigned*)(sortLvl + KCAND); // NPOST
  float*    boxes    = (float*)(sel + NPOST);   // 4*KCAND
  unsigned* mask     = (unsigned*)(boxes + 4 * KCAND); // KCAND*MWORDS (8 MB)

  k_init<<<dim3(64), dim3(256), 0, stream>>>(meta, 32 + 2 * KCAND);
  k_keys<<<dim3((N + 255) / 256), dim3(256), 0, stream>>>(logits, keys, N);
  k_select<<<dim3(NLVL), dim3(1024), 0, stream>>>(keys, levels, N, Tarr, cntG, needE);
  k_compact<<<dim3((N + 255) / 256), dim3(256), 0, stream>>>(keys, levels, N, Tarr, cntG, needE,
                                                             ctrG, ctrE, candKey, candIdx);
  k_sort<<<dim3(1), dim3(1024), 0, stream>>>(candKey, sortedSl);
  k_sepmax<<<dim3((KCAND + 255) / 256), dim3(256), 0, stream>>>(rois, candIdx, sepmax);
  k_shift<<<dim3((KCAND + 255) / 256), dim3(256), 0, stream>>>(rois, candIdx, sortedSl, sepmax,
                                                               boxes, sortIdx, sortLvl);
  k_mask<<<dim3(MWORDS, MWORDS), dim3(32), 0, stream>>>(boxes, mask);
  k_reduce<<<dim3(1), dim3(32), 0, stream>>>(mask, sel, nkeep);
  k_out<<<dim3((NPOST + 255) / 256), dim3(256), 0, stream>>>(logits, rois, sortIdx, sortLvl,
                                                             sel, nkeep, (float*)d_out);
}